// MyModel_90323162235465
// MI455X (gfx1250) — compile-verified
//
#include <hip/hip_runtime.h>

// ---- CDNA5 WMMA fragment types (wave32) ----
typedef __attribute__((ext_vector_type(2))) float v2f;   // A/B frag for 16x16x4 f32
typedef __attribute__((ext_vector_type(8))) float v8f;   // C/D frag (16x16 f32)

static constexpr int FEAT  = 50;   // feature dim
static constexpr int FPAD  = 64;   // padded to multiple of 16 for WMMA tiles
static constexpr int WSTR  = 65;   // LDS stride (floats) for weight tiles, avoids bank conflicts
static constexpr int HSTR  = 65;   // LDS stride for per-wave row-tile staging
static constexpr int WAVES = 8;    // waves per block (256 threads, wave32)
static constexpr int ROWS_PER_WAVE  = 16;
static constexpr int ROWS_PER_BLOCK = WAVES * ROWS_PER_WAVE;   // 128 nodes per block

// ---------------------------------------------------------------------------
// zero-fill
__global__ __launch_bounds__(256) void zero_kernel(float* __restrict__ p, long n) {
    long i = (long)blockIdx.x * blockDim.x + threadIdx.x;
    if (i < n) p[i] = 0.0f;
}

// ---------------------------------------------------------------------------
// Edge scatter-add: agg[dst[e]][f] += feat[src[e]][f]
// 64 threads per edge (lanes 0..49 carry features -> coalesced row reads,
// feature-parallel global_atomic_add_f32 that mostly land in L2: agg = 20MB).
__global__ __launch_bounds__(256) void scatter_kernel(const float* __restrict__ feat,
                                                      const int*  __restrict__ src,
                                                      const int*  __restrict__ dst,
                                                      float* __restrict__ agg,
                                                      int nE) {
    long t = (long)blockIdx.x * blockDim.x + threadIdx.x;
    long e = t >> 6;
    int  f = (int)(t & 63);
    if (e >= nE || f >= FEAT) return;
    int s = src[e];
    int d = dst[e];
    float v = feat[(long)s * FEAT + f];
    unsafeAtomicAdd(&agg[(long)d * FEAT + f], v);   // global_atomic_add_f32, no return
}

// ---------------------------------------------------------------------------
// Fragment helpers for V_WMMA_F32_16X16X4_F32 layouts (ISA 7.12.2):
//   A (16x4):  lane<16 holds row=lane, K={k0,k0+1}; lane>=16 holds row=lane-16, K={k0+2,k0+3}
//   B (4x16):  VGPR0: row K=k0 (lanes 0-15), K=k0+2 (lanes 16-31); VGPR1: K=k0+1 / K=k0+3
//   C/D 16x16: VGPR v holds M=v (lanes 0-15) / M=v+8 (lanes 16-31), N = lane&15 (+n0)

// Fused GIN MLP over a 16-row node tile per wave:
//   h = x + agg  (staged 16x64 in LDS, zero-padded)
//   t = h @ W1^T + b1 ; out = t @ W2^T + b2   (each 16x64x64 via 4x16 WMMAs)
__global__ __launch_bounds__(256) void gin_mlp_kernel(const float* __restrict__ xin,
                                                      const float* __restrict__ agg,
                                                      const float* __restrict__ w1,
                                                      const float* __restrict__ b1,
                                                      const float* __restrict__ w2,
                                                      const float* __restrict__ b2,
                                                      float* __restrict__ hout,
                                                      int nNodes) {
    __shared__ float sW1[FPAD * WSTR];                 // sW1[k][n] = w1[n][k], zero padded
    __shared__ float sW2[FPAD * WSTR];
    __shared__ float sB1[FPAD];
    __shared__ float sB2[FPAD];
    __shared__ float sH[WAVES * ROWS_PER_WAVE * HSTR]; // per-wave row-tile staging

    const int tid = threadIdx.x;

    // Cooperative weight load (transposed into B-matrix orientation, zero pad)
    for (int idx = tid; idx < FPAD * FPAD; idx += 256) {
        int k = idx >> 6, n = idx & 63;
        bool in = (k < FEAT) && (n < FEAT);
        sW1[k * WSTR + n] = in ? w1[n * FEAT + k] : 0.0f;
        sW2[k * WSTR + n] = in ? w2[n * FEAT + k] : 0.0f;
    }
    for (int n = tid; n < FPAD; n += 256) {
        sB1[n] = (n < FEAT) ? b1[n] : 0.0f;
        sB2[n] = (n < FEAT) ? b2[n] : 0.0f;
    }
    __syncthreads();

    const int wave = tid >> 5;
    const int lane = tid & 31;
    const int rowBase = blockIdx.x * ROWS_PER_BLOCK + wave * ROWS_PER_WAVE;
    float* H = &sH[wave * ROWS_PER_WAVE * HSTR];

    // Stage h = x + agg (16 rows x 64 cols, zero padded), coalesced along cols
    for (int i = lane; i < ROWS_PER_WAVE * FPAD; i += 32) {
        int r = i >> 6, c = i & 63;
        long node = (long)rowBase + r;
        float v = 0.0f;
        if (c < FEAT && node < nNodes)
            v = xin[node * FEAT + c] + agg[node * FEAT + c];
        H[r * HSTR + c] = v;
    }
    __syncthreads();

    const int rA     = lane & 15;
    const int koff   = (lane >> 4) << 1;  // 0 for lanes 0-15, 2 for lanes 16-31
    const int rowOff = (lane >> 4) << 3;  // C/D: M = v (+8 for upper half-wave)

    // ---------------- GEMM 1: t = h @ W1^T ----------------
    v8f acc[4] = {v8f{}, v8f{}, v8f{}, v8f{}};
    for (int k0 = 0; k0 < FPAD; k0 += 4) {
        v2f a;
        a.x = H[rA * HSTR + k0 + koff];
        a.y = H[rA * HSTR + k0 + koff + 1];
        const float* Wr0 = &sW1[(k0 + koff) * WSTR];
        const float* Wr1 = &sW1[(k0 + koff + 1) * WSTR];
#pragma unroll
        for (int nt = 0; nt < 4; ++nt) {
            v2f b;
            b.x = Wr0[nt * 16 + rA];
            b.y = Wr1[nt * 16 + rA];
            acc[nt] = __builtin_amdgcn_wmma_f32_16x16x4_f32(
                false, a, false, b, (short)0, acc[nt], false, false);
        }
    }

    // bias add, write t back into staging (padding cols land as exact zeros)
    __syncthreads();
#pragma unroll
    for (int nt = 0; nt < 4; ++nt) {
        int n = nt * 16 + rA;
        float bias = sB1[n];
#pragma unroll
        for (int v = 0; v < 8; ++v)
            H[(v + rowOff) * HSTR + n] = acc[nt][v] + bias;
    }
    __syncthreads();

    // ---------------- GEMM 2: y = t @ W2^T ----------------
#pragma unroll
    for (int nt = 0; nt < 4; ++nt) acc[nt] = v8f{};
    for (int k0 = 0; k0 < FPAD; k0 += 4) {
        v2f a;
        a.x = H[rA * HSTR + k0 + koff];
        a.y = H[rA * HSTR + k0 + koff + 1];
        const float* Wr0 = &sW2[(k0 + koff) * WSTR];
        const float* Wr1 = &sW2[(k0 + koff + 1) * WSTR];
#pragma unroll
        for (int nt = 0; nt < 4; ++nt) {
            v2f b;
            b.x = Wr0[nt * 16 + rA];
            b.y = Wr1[nt * 16 + rA];
            acc[nt] = __builtin_amdgcn_wmma_f32_16x16x4_f32(
                false, a, false, b, (short)0, acc[nt], false, false);
        }
    }

    // bias + store to global (only the 50 valid cols / in-range rows)
#pragma unroll
    for (int nt = 0; nt < 4; ++nt) {
        int n = nt * 16 + rA;
        if (n >= FEAT) continue;
        float bias = sB2[n];
#pragma unroll
        for (int v = 0; v < 8; ++v) {
            long node = (long)rowBase + v + rowOff;
            if (node < nNodes)
                hout[node * FEAT + n] = acc[nt][v] + bias;
        }
    }
}

// ---------------------------------------------------------------------------
// Head: out = relu(h @ l1w^T + l1b) @ l3w^T + l3b   -> [N,1]
__global__ __launch_bounds__(256) void head_kernel(const float* __restrict__ hin,
                                                   const float* __restrict__ l1w,
                                                   const float* __restrict__ l1b,
                                                   const float* __restrict__ l3w,
                                                   const float* __restrict__ l3b,
                                                   float* __restrict__ out,
                                                   int nNodes) {
    __shared__ float sW1[FPAD * WSTR];
    __shared__ float sB1[FPAD];
    __shared__ float sL3[FPAD];
    __shared__ float sH[WAVES * ROWS_PER_WAVE * HSTR];

    const int tid = threadIdx.x;
    for (int idx = tid; idx < FPAD * FPAD; idx += 256) {
        int k = idx >> 6, n = idx & 63;
        bool in = (k < FEAT) && (n < FEAT);
        sW1[k * WSTR + n] = in ? l1w[n * FEAT + k] : 0.0f;
    }
    for (int n = tid; n < FPAD; n += 256) {
        sB1[n] = (n < FEAT) ? l1b[n] : 0.0f;
        sL3[n] = (n < FEAT) ? l3w[n] : 0.0f;
    }
    __syncthreads();

    const int wave = tid >> 5;
    const int lane = tid & 31;
    const int rowBase = blockIdx.x * ROWS_PER_BLOCK + wave * ROWS_PER_WAVE;
    float* H = &sH[wave * ROWS_PER_WAVE * HSTR];

    for (int i = lane; i < ROWS_PER_WAVE * FPAD; i += 32) {
        int r = i >> 6, c = i & 63;
        long node = (long)rowBase + r;
        float v = 0.0f;
        if (c < FEAT && node < nNodes) v = hin[node * FEAT + c];
        H[r * HSTR + c] = v;
    }
    __syncthreads();

    const int rA     = lane & 15;
    const int koff   = (lane >> 4) << 1;
    const int rowOff = (lane >> 4) << 3;

    v8f acc[4] = {v8f{}, v8f{}, v8f{}, v8f{}};
    for (int k0 = 0; k0 < FPAD; k0 += 4) {
        v2f a;
        a.x = H[rA * HSTR + k0 + koff];
        a.y = H[rA * HSTR + k0 + koff + 1];
        const float* Wr0 = &sW1[(k0 + koff) * WSTR];
        const float* Wr1 = &sW1[(k0 + koff + 1) * WSTR];
#pragma unroll
        for (int nt = 0; nt < 4; ++nt) {
            v2f b;
            b.x = Wr0[nt * 16 + rA];
            b.y = Wr1[nt * 16 + rA];
            acc[nt] = __builtin_amdgcn_wmma_f32_16x16x4_f32(
                false, a, false, b, (short)0, acc[nt], false, false);
        }
    }

    // bias + ReLU, stash t into staging
    __syncthreads();
#pragma unroll
    for (int nt = 0; nt < 4; ++nt) {
        int n = nt * 16 + rA;
        float bias = sB1[n];
#pragma unroll
        for (int v = 0; v < 8; ++v) {
            float t = acc[nt][v] + bias;
            H[(v + rowOff) * HSTR + n] = t > 0.0f ? t : 0.0f;
        }
    }
    __syncthreads();

    // final 50->1 dot per row (lanes 0-15 each own one row)
    if (lane < ROWS_PER_WAVE) {
        long node = (long)rowBase + lane;
        if (node < nNodes) {
            float s = 0.0f;
            const float* row = &H[lane * HSTR];
#pragma unroll
            for (int n = 0; n < FEAT; ++n) s += row[n] * sL3[n];
            out[node] = s + l3b[0];
        }
    }
}

// ---------------------------------------------------------------------------
extern "C" void kernel_launch(void* const* d_in, const int* in_sizes, int n_in,
                              void* d_out, int out_size, void* d_ws, size_t ws_size,
                              hipStream_t stream) {
    const float* x    = (const float*)d_in[0];
    const int*   ei   = (const int*)d_in[1];
    const float* c1w1 = (const float*)d_in[2];
    const float* c1b1 = (const float*)d_in[3];
    const float* c1w2 = (const float*)d_in[4];
    const float* c1b2 = (const float*)d_in[5];
    const float* c2w1 = (const float*)d_in[6];
    const float* c2b1 = (const float*)d_in[7];
    const float* c2w2 = (const float*)d_in[8];
    const float* c2b2 = (const float*)d_in[9];
    const float* l1w  = (const float*)d_in[10];
    const float* l1b  = (const float*)d_in[11];
    const float* l3w  = (const float*)d_in[12];
    const float* l3b  = (const float*)d_in[13];

    const int nN = in_sizes[0] / FEAT;
    const int nE = in_sizes[1] / 2;
    const int* src = ei;
    const int* dst = ei + nE;

    // workspace: agg (reused), h1, h2 -- each N*50 floats, 256B-aligned slabs
    const long   nFeatElems = (long)nN * FEAT;
    const size_t slab = ((size_t)nFeatElems * sizeof(float) + 255) & ~(size_t)255;
    float* agg = (float*)d_ws;
    float* h1  = (float*)((char*)d_ws + slab);
    float* h2  = (float*)((char*)d_ws + 2 * slab);
    float* out = (float*)d_out;

    const int zeroBlocks    = (int)((nFeatElems + 255) / 256);
    const int scatterBlocks = (int)(((long)nE * 64 + 255) / 256);
    const int tileBlocks    = (nN + ROWS_PER_BLOCK - 1) / ROWS_PER_BLOCK;

    // ---- layer 1 ----
    zero_kernel<<<zeroBlocks, 256, 0, stream>>>(agg, nFeatElems);
    scatter_kernel<<<scatterBlocks, 256, 0, stream>>>(x, src, dst, agg, nE);
    gin_mlp_kernel<<<tileBlocks, 256, 0, stream>>>(x, agg, c1w1, c1b1, c1w2, c1b2, h1, nN);

    // ---- layer 2 ----
    zero_kernel<<<zeroBlocks, 256, 0, stream>>>(agg, nFeatElems);
    scatter_kernel<<<scatterBlocks, 256, 0, stream>>>(h1, src, dst, agg, nE);
    gin_mlp_kernel<<<tileBlocks, 256, 0, stream>>>(h1, agg, c2w1, c2b1, c2w2, c2b2, h2, nN);

    // ---- head ----
    head_kernel<<<tileBlocks, 256, 0, stream>>>(h2, l1w, l1b, l3w, l3b, out, nN);
}